// PostProcess_65352222376170
// MI455X (gfx1250) — compile-verified
//
#include <hip/hip_runtime.h>
#include <stdint.h>

// ---------------------------------------------------------------------------
// DETR-style post-processing for MI455X (gfx1250, wave32).
//
// Key optimizations (reasoned from MI455X specs, no timing available):
//  * Never read the 512MB pred_adj tensor fully: gather only the 100x100
//    entries per batch selected by top-k (1.28MB instead of 512MB of HBM).
//  * sigmoid is monotone -> top-k over raw logits; sigmoid only the winners.
//  * Exact radix-select (4x 256-bin passes) + bitonic sort reproduces
//    lax.top_k ordering (desc value, ties by ascending index).
//  * Logits (2x23MB) are L2-resident (192MB L2), so re-scan passes are cheap.
//  * CDNA5 paths: Tensor Data Mover stages pass-0 chunks into LDS
//    (tensor_load_to_lds + s_wait_tensorcnt), global_prefetch_b8 on scans.
// ---------------------------------------------------------------------------

#define B_  32
#define Q_  2000
#define C_  91
#define A_  18
#define K_  100
#define N_  (Q_ * C_)        // 182000 logits per (batch, tensor)

#define NT   1024            // 32 wave32s per selection workgroup
#define CAP  2048            // candidate buffer (power of two for bitonic)
#define CH   8192            // staged chunk elements (32KB LDS)

typedef unsigned int v4u __attribute__((ext_vector_type(4)));
typedef int          v8i __attribute__((ext_vector_type(8)));
typedef int          v4i __attribute__((ext_vector_type(4)));

#if defined(__gfx1250__) && __has_builtin(__builtin_amdgcn_tensor_load_to_lds) && __has_builtin(__builtin_amdgcn_s_wait_tensorcnt)
#define USE_TDM 1
#else
#define USE_TDM 0
#endif

// Monotone float -> uint key (larger float => larger key).
__device__ __forceinline__ unsigned f2ord(float f) {
  unsigned u = __float_as_uint(f);
  return (u & 0x80000000u) ? ~u : (u | 0x80000000u);
}
__device__ __forceinline__ float ord2f(unsigned k) {
  unsigned u = (k & 0x80000000u) ? (k & 0x7fffffffu) : ~k;
  return __uint_as_float(u);
}
__device__ __forceinline__ float sigmoidf(float x) {
  return 1.0f / (1.0f + __expf(-x));
}

// ---------------------------------------------------------------------------
// Kernel 1: exact top-K selection per (tensor, batch).
// grid = (32, 2): blockIdx.x = batch, blockIdx.y = 0:pred_logits 1:grasp.
// Writes sel_val (sigmoid of winning logit) and sel_idx (flat index in Q*C),
// sorted descending by value with jax tie-breaking (lower index first).
// ---------------------------------------------------------------------------
__global__ __launch_bounds__(NT) void topk_select_kernel(
    const float* __restrict__ logits,
    const float* __restrict__ logits_g,
    float* __restrict__ sel_val,
    int*   __restrict__ sel_idx) {
  const int b   = blockIdx.x;
  const int t   = blockIdx.y;
  const int tid = threadIdx.x;
  const float* p = (t == 0 ? logits : logits_g) + (size_t)b * N_;

  __shared__ unsigned hist[256];
  __shared__ unsigned s_prefix;
  __shared__ int      s_kth;
  __shared__ unsigned s_cnt;
  __shared__ float    stage[CH];
  __shared__ unsigned skey[CAP];
  __shared__ int      sidx[CAP];

  if (tid == 0) { s_prefix = 0u; s_kth = K_; s_cnt = 0u; }
  for (int i = tid; i < 256; i += NT) hist[i] = 0u;
  __syncthreads();

  // ---- pass 0: histogram of top byte; data staged into LDS via TDM -------
  for (int base = 0; base < N_; base += CH) {
    const int len = (N_ - base < CH) ? (N_ - base) : CH;
#if USE_TDM
    if (tid < 32) {  // wave 0 issues the Tensor Data Mover op
      const size_t   ga  = (size_t)(const void*)(p + base);
      const unsigned lds = (unsigned)(size_t)(void*)stage;
      const unsigned td0 = (unsigned)len;
      v4u g0;
      g0.x = 1u;                                   // count=1 (valid), load
      g0.y = lds;                                  // lds_addr (bytes)
      g0.z = (unsigned)(ga & 0xffffffffu);         // global_addr[31:0]
      g0.w = (unsigned)(ga >> 32) | (2u << 30);    // global_addr[56:32], type=2
      v8i g1;
      g1[0] = (int)(2u << 16);                     // wg_mask=0, data_size=4B
      g1[1] = (int)((td0 & 0xffffu) << 16);        // tensor_dim0[15:0]
      g1[2] = (int)((td0 >> 16) | (1u << 16));     // tensor_dim0[31:16], tensor_dim1=1
      g1[3] = (int)((td0 & 0xffffu) << 16);        // tile_dim0 = len
      g1[4] = (int)(1u | (1u << 16));              // tile_dim1=1, tile_dim2=1
      g1[5] = (int)td0;                            // tensor_dim0_stride
      g1[6] = 0;
      g1[7] = 0;
      v4i g2 = {0, 0, 0, 0};
      v4i g3 = {0, 0, 0, 0};
      v8i g4 = {0, 0, 0, 0, 0, 0, 0, 0};
      __builtin_amdgcn_tensor_load_to_lds(g0, g1, g2, g3, g4, 0);
      __builtin_amdgcn_s_wait_tensorcnt(0);
    }
    __syncthreads();
#else
    for (int i = tid; i < len; i += NT) stage[i] = p[base + i];
    __syncthreads();
#endif
    for (int i = tid; i < len; i += NT) {
      unsigned k = f2ord(stage[i]);
      atomicAdd(&hist[k >> 24], 1u);
    }
    __syncthreads();
  }
  if (tid == 0) {
    unsigned cum = 0u;
    for (int bin = 255; bin >= 0; --bin) {
      unsigned c = hist[bin];
      if (cum + c >= (unsigned)s_kth) { s_prefix |= (unsigned)bin << 24; s_kth -= (int)cum; break; }
      cum += c;
    }
  }
  __syncthreads();

  // ---- passes 1..3: refine threshold byte by byte (L2-resident rescans) ---
  for (int pass = 1; pass <= 3; ++pass) {
    const int shift = 24 - 8 * pass;
    for (int i = tid; i < 256; i += NT) hist[i] = 0u;
    __syncthreads();
    const unsigned hi = s_prefix >> (shift + 8);
    for (int i = tid; i < N_; i += NT) {
      if (i + 4 * NT < N_) __builtin_prefetch(p + i + 4 * NT, 0, 1);
      unsigned k = f2ord(p[i]);
      if ((k >> (shift + 8)) == hi) atomicAdd(&hist[(k >> shift) & 255u], 1u);
    }
    __syncthreads();
    if (tid == 0) {
      unsigned cum = 0u;
      for (int bin = 255; bin >= 0; --bin) {
        unsigned c = hist[bin];
        if (cum + c >= (unsigned)s_kth) { s_prefix |= (unsigned)bin << shift; s_kth -= (int)cum; break; }
        cum += c;
      }
    }
    __syncthreads();
  }

  // ---- collect all elements with key >= exact threshold -------------------
  const unsigned thr = s_prefix;   // exact K-th largest key
  for (int i = tid; i < CAP; i += NT) { skey[i] = 0u; sidx[i] = 0x7fffffff; }
  __syncthreads();
  for (int i = tid; i < N_; i += NT) {
    if (i + 4 * NT < N_) __builtin_prefetch(p + i + 4 * NT, 0, 1);
    unsigned k = f2ord(p[i]);
    if (k >= thr) {
      unsigned pos = atomicAdd(&s_cnt, 1u);
      if (pos < CAP) { skey[pos] = k; sidx[pos] = i; }
    }
  }
  __syncthreads();

  // ---- bitonic sort (key desc, index asc) — canonical, deterministic ------
  for (int k = 2; k <= CAP; k <<= 1) {
    for (int j = k >> 1; j > 0; j >>= 1) {
      for (int i = tid; i < CAP; i += NT) {
        int l = i ^ j;
        if (l > i) {
          bool up = ((i & k) == 0);
          unsigned ki = skey[i], kl = skey[l];
          int ii = sidx[i], il = sidx[l];
          bool iBefore = (ki > kl) || (ki == kl && ii < il);
          if (up ? !iBefore : iBefore) {
            skey[i] = kl; skey[l] = ki;
            sidx[i] = il; sidx[l] = ii;
          }
        }
      }
      __syncthreads();
    }
  }

  // ---- emit winners --------------------------------------------------------
  const int tb = t * B_ + b;
  if (tid < K_) {
    sel_val[tb * K_ + tid] = sigmoidf(ord2f(skey[tid]));
    sel_idx[tb * K_ + tid] = sidx[tid];
  }
}

// ---------------------------------------------------------------------------
// Kernel 2: per-batch finalize — labels/keep, box transforms, grasp boxes +
// angle argmax, and the sparse 100x100 adj gather (+sigmoid, pair mask).
// ---------------------------------------------------------------------------
__global__ __launch_bounds__(256) void finalize_kernel(
    const float* __restrict__ pred_boxes,
    const float* __restrict__ pred_adj,
    const float* __restrict__ pred_angles_grasp,
    const float* __restrict__ pred_boxes_grasp,
    const float* __restrict__ target_sizes,
    const float* __restrict__ sel_val,
    const int*   __restrict__ sel_idx,
    float* __restrict__ out) {
  const int b   = blockIdx.x;
  const int tid = threadIdx.x;

  // Flat output layout, tuple return order:
  float* o_topkv   = out;             // (32,100)        [0,      3200)
  float* o_labels  = out + 3200;      // (32,100)        [3200,   6400)
  float* o_boxes   = out + 6400;      // (32,100,4)      [6400,  19200)
  float* o_topkvg  = out + 19200;     // (32,100)        [19200, 22400)
  float* o_labelsg = out + 22400;     // (32,100)        [22400, 25600)
  float* o_boxesg  = out + 25600;     // (32,100,5)      [25600, 41600)
  float* o_adj     = out + 41600;     // (32,100,100)    [41600, 361600)
  float* o_keep    = out + 361600;    // (32,100)        [361600,364800)

  __shared__ int s_tb[K_];
  __shared__ int s_keep[K_];

  const float img_h = target_sizes[b * 2 + 0];
  const float img_w = target_sizes[b * 2 + 1];

  if (tid < K_) {
    // detection stream
    float v  = sel_val[b * K_ + tid];
    int idx  = sel_idx[b * K_ + tid];
    int tb   = idx / C_;
    int lab  = idx - tb * C_;
    int keep = (v > 0.3f) ? 1 : 0;
    s_tb[tid]   = tb;
    s_keep[tid] = keep;

    o_topkv [b * K_ + tid] = v;
    o_labels[b * K_ + tid] = keep ? (float)lab : -1.0f;
    o_keep  [b * K_ + tid] = keep ? 1.0f : 0.0f;

    const float* bx = pred_boxes + ((size_t)b * Q_ + tb) * 4;
    float cx = bx[0], cy = bx[1], w = bx[2], h = bx[3];
    float* ob = o_boxes + ((size_t)b * K_ + tid) * 4;
    ob[0] = (cx - 0.5f * w) * img_w;
    ob[1] = (cy - 0.5f * h) * img_h;
    ob[2] = (cx + 0.5f * w) * img_w;
    ob[3] = (cy + 0.5f * h) * img_h;

    // grasp stream
    float vg = sel_val[(B_ + b) * K_ + tid];
    int idxg = sel_idx[(B_ + b) * K_ + tid];
    int tbg  = idxg / C_;
    int labg = idxg - tbg * C_;
    o_topkvg [b * K_ + tid] = vg;
    o_labelsg[b * K_ + tid] = (float)labg;

    // argmax over 18 angle bins (sigmoid monotone -> argmax on raw logits;
    // strict > keeps jnp.argmax first-occurrence tie-breaking)
    const float* ag = pred_angles_grasp + ((size_t)b * Q_ + tbg) * A_;
    int amax = 0; float abest = ag[0];
    for (int a = 1; a < A_; ++a) {
      float x = ag[a];
      if (x > abest) { abest = x; amax = a; }
    }
    float angle = (float)((amax - 8) * 10 - 5);

    const float* bg = pred_boxes_grasp + ((size_t)b * Q_ + tbg) * 4;
    float* obg = o_boxesg + ((size_t)b * K_ + tid) * 5;
    obg[0] = bg[0] * img_w;
    obg[1] = bg[1] * img_h;
    obg[2] = bg[2] * img_w;
    obg[3] = bg[3] * img_h;
    obg[4] = angle;
  }
  __syncthreads();

  // Sparse adjacency gather: 10000 scattered 4B reads instead of 16MB/batch.
  const float* adjb = pred_adj + (size_t)b * Q_ * Q_;
  for (int p = tid; p < K_ * K_; p += 256) {
    int i = p / K_;
    int j = p - i * K_;
    float v = 0.0f;
    if (s_keep[i] && s_keep[j]) {
      v = sigmoidf(adjb[(size_t)s_tb[i] * Q_ + s_tb[j]]);
    }
    o_adj[(size_t)b * (K_ * K_) + p] = v;
  }
}

// ---------------------------------------------------------------------------
extern "C" void kernel_launch(void* const* d_in, const int* in_sizes, int n_in,
                              void* d_out, int out_size, void* d_ws, size_t ws_size,
                              hipStream_t stream) {
  const float* pred_logits       = (const float*)d_in[0];
  const float* pred_boxes        = (const float*)d_in[1];
  const float* pred_adj          = (const float*)d_in[2];
  const float* pred_logits_grasp = (const float*)d_in[3];
  const float* pred_angles_grasp = (const float*)d_in[4];
  const float* pred_boxes_grasp  = (const float*)d_in[5];
  const float* target_sizes      = (const float*)d_in[6];
  float* out = (float*)d_out;

  // Workspace: sel_val[2][32][100] floats, sel_idx[2][32][100] ints (51200 B).
  float* sel_val = (float*)d_ws;
  int*   sel_idx = (int*)((char*)d_ws + (size_t)2 * B_ * K_ * sizeof(float));

  dim3 gsel(B_, 2);
  topk_select_kernel<<<gsel, NT, 0, stream>>>(pred_logits, pred_logits_grasp,
                                              sel_val, sel_idx);
  finalize_kernel<<<B_, 256, 0, stream>>>(pred_boxes, pred_adj,
                                          pred_angles_grasp, pred_boxes_grasp,
                                          target_sizes, sel_val, sel_idx, out);
}